// ScaledDotProductAttention_49959059587084
// MI455X (gfx1250) — compile-verified
//
#include <hip/hip_runtime.h>
#include <hip/hip_bf16.h>
#include <math.h>

typedef __attribute__((ext_vector_type(2))) float v2f;
typedef __attribute__((ext_vector_type(8))) float v8f;

#define B_ 32
#define L_ 1024
#define D_ 64
#define J_ (2 * L_ - 1)

static __device__ __forceinline__ v8f wmma_f32(v2f a, v2f b, v8f c) {
    return __builtin_amdgcn_wmma_f32_16x16x4_f32(
        /*neg_a=*/false, a, /*neg_b=*/false, b,
        /*c_mod=*/(short)0, c, /*reuse_a=*/false, /*reuse_b=*/false);
}

static __device__ __forceinline__ float bperm(int srcLane, float v) {
    return __int_as_float(__builtin_amdgcn_ds_bpermute(srcLane << 2, __float_as_int(v)));
}

// ---------------------------------------------------------------------------
// Fused kernel: logits (QK^T + positional stripe, mask) + row softmax,
// writing the final softmaxed attention exactly once.
// Block = 256 threads (8 waves). Each block owns a 16-row x 1024-col panel.
// Wave w computes 8 column tiles; logits live entirely in registers.
// ---------------------------------------------------------------------------
__global__ __launch_bounds__(256) void attn_kernel(
    const float* __restrict__ q,
    const float* __restrict__ k,
    const unsigned char* __restrict__ mask,
    const float* __restrict__ pd,   // [B, 2L-1, D]
    const float* __restrict__ qp,   // [B, L, D]
    float* __restrict__ attn)       // [B, L, L]
{
    const int t0   = blockIdx.x * 16;
    const int b    = blockIdx.y;
    const int tid  = threadIdx.x;
    const int w    = tid >> 5;       // wave 0..7
    const int lane = tid & 31;
    const int hi   = lane >> 4;      // 0/1
    const int lr   = lane & 15;

    const float* kb  = k  + (size_t)b * L_ * D_;
    const float* pdb = pd + (size_t)b * J_ * D_;

    // Hoist q / q_position A-fragments: identical for all 8 column tiles.
    const float* qrow  = q  + ((size_t)b * L_ + (t0 + lr)) * D_ + 2 * hi;
    const float* qprow = qp + ((size_t)b * L_ + (t0 + lr)) * D_ + 2 * hi;
    v2f aq[16], ap[16];
#pragma unroll
    for (int kk = 0; kk < 16; ++kk) {
        aq[kk] = *(const v2f*)(qrow  + 4 * kk);
        ap[kk] = *(const v2f*)(qprow + 4 * kk);
    }

    const unsigned char* maskp = mask + ((size_t)b * L_ + t0) * L_;
    const float rtemper = 0.125f;    // 1/sqrt(64)

    v8f e[8];                        // logits, later exp values, for 8 tiles
#pragma unroll
    for (int i = 0; i < 8; ++i) {
        const int m0 = (w * 8 + i) * 16;
        const int J0 = (L_ - 1) - t0 + m0;          // band [J0-15, J0+15] always in range
        const float* krow = kb  + (size_t)(m0 + lr)      * D_ + 2 * hi;
        const float* pdl  = pdb + (size_t)(J0 - 15 + lr) * D_ + 2 * hi;
        const float* pdr  = pdb + (size_t)(J0 + lr)      * D_ + 2 * hi;

        v8f acc = {}, gl = {}, gr = {};
#pragma unroll
        for (int kk = 0; kk < 16; ++kk) {
            acc = wmma_f32(aq[kk], *(const v2f*)(krow + 4 * kk), acc);
            gl  = wmma_f32(ap[kk], *(const v2f*)(pdl  + 4 * kk), gl);
            gr  = wmma_f32(ap[kk], *(const v2f*)(pdr  + 4 * kk), gr);
        }

        // Diagonal-band gather via cross-lane permute (no LDS, no barrier):
        // G[dt, c] lives in register (dt&7) of lane c + 16*(dt>=8).
        v8f val;
#pragma unroll
        for (int r = 0; r < 8; ++r) {
            const int dt   = r + 8 * hi;
            const int diff = lr - dt;                         // dm - dt
            const float gR = bperm(((diff      + 16 * hi) & 31), gr[r]); // c = diff
            const float gL = bperm(((15 + diff + 16 * hi) & 31), gl[r]); // c = 15+diff
            const float pos = (diff >= 0) ? gR : gL;
            float x = (acc[r] + pos) * rtemper;
            if (maskp[(size_t)dt * L_ + m0 + lr]) x = -INFINITY;
            val[r] = x;
        }
        e[i] = val;
    }

    // ---- row softmax across the whole panel (rows dt = r + 8*hi) ----
    __shared__ float sMax[8][16];
    __shared__ float sSum[8][16];

    float rmax[8];
#pragma unroll
    for (int r = 0; r < 8; ++r) {
        float m = e[0][r];
#pragma unroll
        for (int i = 1; i < 8; ++i) m = fmaxf(m, e[i][r]);
#pragma unroll
        for (int d = 1; d < 16; d <<= 1) m = fmaxf(m, __shfl_xor(m, d, 32));
        rmax[r] = m;
    }
    if (lr == 0) {
#pragma unroll
        for (int r = 0; r < 8; ++r) sMax[w][r + 8 * hi] = rmax[r];
    }
    __syncthreads();
#pragma unroll
    for (int r = 0; r < 8; ++r) {
        const int dt = r + 8 * hi;
        float m = sMax[0][dt];
#pragma unroll
        for (int ww = 1; ww < 8; ++ww) m = fmaxf(m, sMax[ww][dt]);
        rmax[r] = m;
    }

    float rinv[8];
#pragma unroll
    for (int r = 0; r < 8; ++r) {
        float s = 0.f;
#pragma unroll
        for (int i = 0; i < 8; ++i) {
            const float ex = __expf(e[i][r] - rmax[r]);
            e[i][r] = ex;
            s += ex;
        }
#pragma unroll
        for (int d = 1; d < 16; d <<= 1) s += __shfl_xor(s, d, 32);
        rinv[r] = s;
    }
    if (lr == 0) {
#pragma unroll
        for (int r = 0; r < 8; ++r) sSum[w][r + 8 * hi] = rinv[r];
    }
    __syncthreads();
#pragma unroll
    for (int r = 0; r < 8; ++r) {
        const int dt = r + 8 * hi;
        float s = sSum[0][dt];
#pragma unroll
        for (int ww = 1; ww < 8; ++ww) s += sSum[ww][dt];
        rinv[r] = 1.0f / s;
    }

    // ---- single write of the final attention panel ----
    float* arow = attn + ((size_t)b * L_ + t0) * L_;
#pragma unroll
    for (int i = 0; i < 8; ++i) {
        const int m0 = (w * 8 + i) * 16;
#pragma unroll
        for (int r = 0; r < 8; ++r) {
            const int dt = r + 8 * hi;
            arow[(size_t)dt * L_ + m0 + lr] = e[i][r] * rinv[r];
        }
    }
}

// ---------------------------------------------------------------------------
// Kernel 2: output = attn @ V.  One wave per 16x16 output tile, K = L.
// ---------------------------------------------------------------------------
__global__ void pv_kernel(const float* __restrict__ attn,   // [B, L, L]
                          const float* __restrict__ v,      // [B, L, D]
                          float* __restrict__ out)          // [B, L, D]
{
    const int n0   = blockIdx.x * 16;   // 0..D-16
    const int t0   = blockIdx.y * 16;
    const int b    = blockIdx.z;
    const int lane = threadIdx.x;
    const int hi   = lane >> 4;
    const int lr   = lane & 15;

    const float* arow = attn + ((size_t)b * L_ + (t0 + lr)) * L_ + 2 * hi;
    const float* vb   = v + (size_t)b * L_ * D_ + n0 + lr;

    v8f acc = {};
#pragma unroll 8
    for (int kk = 0; kk < L_ / 4; ++kk) {
        v2f a = *(const v2f*)(arow + 4 * kk);
        const int kbase = 4 * kk + 2 * hi;
        v2f bf;
        bf.x = vb[(size_t)(kbase + 0) * D_];
        bf.y = vb[(size_t)(kbase + 1) * D_];
        acc = wmma_f32(a, bf, acc);
    }

#pragma unroll
    for (int r = 0; r < 8; ++r) {
        out[((size_t)b * L_ + (t0 + r + 8 * hi)) * D_ + n0 + lr] = acc[r];
    }
}

// ---------------------------------------------------------------------------
extern "C" void kernel_launch(void* const* d_in, const int* in_sizes, int n_in,
                              void* d_out, int out_size, void* d_ws, size_t ws_size,
                              hipStream_t stream)
{
    (void)in_sizes; (void)n_in; (void)out_size; (void)d_ws; (void)ws_size;

    const float*         q    = (const float*)d_in[0];
    const float*         k    = (const float*)d_in[1];
    const float*         v    = (const float*)d_in[2];
    const unsigned char* mask = (const unsigned char*)d_in[3];   // jnp bool -> 1 byte
    const float*         pd   = (const float*)d_in[4];
    const float*         qp   = (const float*)d_in[5];

    float* out  = (float*)d_out;                       // [B, L, D]
    float* attn = out + (size_t)B_ * L_ * D_;          // [B, L, L]

    dim3 gridA(L_ / 16, B_);
    attn_kernel<<<gridA, dim3(256), 0, stream>>>(q, k, mask, pd, qp, attn);

    dim3 gridC(D_ / 16, L_ / 16, B_);
    pv_kernel<<<gridC, dim3(32), 0, stream>>>(attn, v, out);
}